// LagAlignPro_81527069213319
// MI455X (gfx1250) — compile-verified
//
#include <hip/hip_runtime.h>
#include <math.h>

typedef __attribute__((ext_vector_type(2))) float v2f;
typedef __attribute__((ext_vector_type(8))) float v8f;

#define TT     512
#define DD     32
#define HID    64
#define CHUNK  128
#define HSTR   65     // padded LDS row stride for H (odd -> conflict-free column reads)
#define W2STR  33     // padded LDS row stride for k_w2

__launch_bounds__(256, 2)
__global__ void lagalign_fused_kernel(const float* __restrict__ X,
                                      const float* __restrict__ k_w1, const float* __restrict__ k_b1,
                                      const float* __restrict__ k_w2, const float* __restrict__ k_b2,
                                      const float* __restrict__ q_w1, const float* __restrict__ q_b1,
                                      const float* __restrict__ q_w2, const float* __restrict__ q_b2,
                                      const float* __restrict__ pos_bias,
                                      float* __restrict__ Yout, float* __restrict__ Wout)
{
    __shared__ float Xs[TT];
    __shared__ float Hs[CHUNK * HSTR];
    __shared__ float w1s[3 * HID];
    __shared__ float b1s[HID];
    __shared__ float w2s[HID * W2STR];
    __shared__ float b2s[DD];
    __shared__ float qs[DD];
    __shared__ float hqs[HID];
    __shared__ float logitsS[TT];
    __shared__ float red[8];
    __shared__ float red2[8];
    __shared__ float scal[1];

    const int tid  = threadIdx.x;
    const int lane = tid & 31;
    const int wid  = tid >> 5;
    const int bv   = blockIdx.x;
    const float* Xr = X + (size_t)bv * TT;

    // ---- stage X row and K-head weights in LDS ----
    for (int i = tid; i < TT; i += 256)      Xs[i]  = Xr[i];
    for (int i = tid; i < 3 * HID; i += 256) w1s[i] = k_w1[i];
    if (tid < HID) b1s[tid] = k_b1[tid];
    for (int i = tid; i < HID * DD; i += 256)
        w2s[(i >> 5) * W2STR + (i & 31)] = k_w2[i];
    if (tid < DD) b2s[tid] = k_b2[tid];
    __syncthreads();

    // ---- query head: q = relu([X509,X510,X511] @ q_w1 + q_b1) @ q_w2 + q_b2 ----
    if (tid < HID) {
        float p0 = Xs[TT - 3], p1 = Xs[TT - 2], p2 = Xs[TT - 1];
        float a = p0 * q_w1[0 * HID + tid] + p1 * q_w1[1 * HID + tid]
                + p2 * q_w1[2 * HID + tid] + q_b1[tid];
        hqs[tid] = a > 0.f ? a : 0.f;
    }
    __syncthreads();
    if (tid < DD) {
        float acc = q_b2[tid];
        for (int h = 0; h < HID; ++h) acc += hqs[h] * q_w2[h * DD + tid];
        qs[tid] = acc;
    }
    __syncthreads();

    const float inv_sqrt_d = 0.17677669529663687f; // 1/sqrt(32)

    for (int chunk = 0; chunk < TT / CHUNK; ++chunk) {
        const int tbase = chunk * CHUNK;

        // ---- phase 1: H[tl][h] = relu(3-tap conv) into LDS ----
        for (int idx = tid; idx < CHUNK * HID; idx += 256) {
            int tl = idx >> 6;           // / HID
            int h  = idx & (HID - 1);
            int t  = tbase + tl;
            int t0 = (t - 2 < 0) ? 0 : t - 2;
            int t1 = (t - 1 < 0) ? 0 : t - 1;
            float a = Xs[t0] * w1s[h] + Xs[t1] * w1s[HID + h]
                    + Xs[t]  * w1s[2 * HID + h] + b1s[h];
            Hs[tl * HSTR + h] = a > 0.f ? a : 0.f;
        }
        __syncthreads();

        // ---- phase 2: K-tile = H(16x64) @ k_w2(64x32) + b2 via f32 WMMA, then K.q ----
        {
            const int m    = lane & 15;
            const int half = lane >> 4;   // 0: K-pair {0,1}; 1: K-pair {2,3}
            const int koff = half * 2;
            const int rowb = wid * 16;    // this wave's 16 t-rows inside the chunk

            v8f c0, c1;                   // accumulators for d=0..15 and d=16..31
            float bc0 = b2s[m], bc1 = b2s[16 + m];
            for (int r = 0; r < 8; ++r) { c0[r] = bc0; c1[r] = bc1; }

            for (int kk = 0; kk < HID / 4; ++kk) {
                int kb = kk * 4 + koff;
                v2f a, b0, b1v;
                // A: 16x4 f32 tile of H (ISA layout: lanes 0-15 K={0,1}, 16-31 K={2,3})
                a[0]  = Hs[(rowb + m) * HSTR + kb];
                a[1]  = Hs[(rowb + m) * HSTR + kb + 1];
                // B: 4x16 f32 tiles of k_w2 (mirrored layout, N = lane)
                b0[0]  = w2s[kb * W2STR + m];
                b0[1]  = w2s[(kb + 1) * W2STR + m];
                b1v[0] = w2s[kb * W2STR + 16 + m];
                b1v[1] = w2s[(kb + 1) * W2STR + 16 + m];
                c0 = __builtin_amdgcn_wmma_f32_16x16x4_f32(false, a, false, b0,
                                                           (short)0, c0, false, false);
                c1 = __builtin_amdgcn_wmma_f32_16x16x4_f32(false, a, false, b1v,
                                                           (short)0, c1, false, false);
            }

            // logits[m] = sum_d K[m][d] * q[d]  (lane index == d in the C layout)
            float qa = qs[m], qb = qs[16 + m];
            float part[8];
            for (int r = 0; r < 8; ++r) part[r] = c0[r] * qa + c1[r] * qb;
            for (int mask = 1; mask <= 8; mask <<= 1)
                for (int r = 0; r < 8; ++r)
                    part[r] += __shfl_xor(part[r], mask, 32);

            if (m == 0) { // lanes 0 (rows 0-7) and 16 (rows 8-15) hold the sums
                for (int r = 0; r < 8; ++r) {
                    int t = tbase + rowb + half * 8 + r;
                    float l = part[r] * inv_sqrt_d + pos_bias[t];
                    l = (t <= TT - 2) ? l : -1.0e9f;   // only t = T-1 is masked
                    logitsS[t] = l / 0.7f;
                }
            }
        }
        __syncthreads();
    }

    // ---- phase 3: softmax over T, W and Y ----
    float l0 = logitsS[tid], l1 = logitsS[tid + 256];
    float lm = fmaxf(l0, l1);
    for (int mask = 16; mask >= 1; mask >>= 1) lm = fmaxf(lm, __shfl_xor(lm, mask, 32));
    if (lane == 0) red[wid] = lm;
    __syncthreads();
    if (tid == 0) {
        float M = red[0];
        for (int i = 1; i < 8; ++i) M = fmaxf(M, red[i]);
        scal[0] = M;
    }
    __syncthreads();
    float M = scal[0];
    float e0 = __expf(l0 - M), e1 = __expf(l1 - M);
    float ssum = e0 + e1;
    float ysum = e0 * Xs[tid] + e1 * Xs[tid + 256];
    for (int mask = 16; mask >= 1; mask >>= 1) {
        ssum += __shfl_xor(ssum, mask, 32);
        ysum += __shfl_xor(ysum, mask, 32);
    }
    __syncthreads();   // make sure everyone read M before scal is reused
    if (lane == 0) { red[wid] = ssum; red2[wid] = ysum; }
    __syncthreads();
    if (tid == 0) {
        float S = 0.f, Y = 0.f;
        for (int i = 0; i < 8; ++i) { S += red[i]; Y += red2[i]; }
        scal[0] = S;
        Yout[bv] = Y / S;
    }
    __syncthreads();
    float S = scal[0];
    float* Wr = Wout + (size_t)bv * TT;
    Wr[tid]       = e0 / S;
    Wr[tid + 256] = e1 / S;
}

extern "C" void kernel_launch(void* const* d_in, const int* in_sizes, int n_in,
                              void* d_out, int out_size, void* d_ws, size_t ws_size,
                              hipStream_t stream) {
    const float* X    = (const float*)d_in[0];
    const float* k_w1 = (const float*)d_in[1];
    const float* k_b1 = (const float*)d_in[2];
    const float* k_w2 = (const float*)d_in[3];
    const float* k_b2 = (const float*)d_in[4];
    const float* q_w1 = (const float*)d_in[5];
    const float* q_b1 = (const float*)d_in[6];
    const float* q_w2 = (const float*)d_in[7];
    const float* q_b2 = (const float*)d_in[8];
    const float* pos  = (const float*)d_in[9];

    float* Yout = (float*)d_out;            // (B*V) = 4096 floats
    float* Wout = Yout + 32 * 128;          // (B*V*T) floats follow

    dim3 grid(32 * 128);                    // one block per (b,v) row
    dim3 block(256);                        // 8 waves of 32
    lagalign_fused_kernel<<<grid, block, 0, stream>>>(
        X, k_w1, k_b1, k_w2, k_b2, q_w1, q_b1, q_w2, q_b2, pos, Yout, Wout);
}